// ExpertRouter_86835648790910
// MI455X (gfx1250) — compile-verified
//
#include <hip/hip_runtime.h>
#include <hip/hip_bf16.h>

#define NUM_EXP 8
#define N_EMB   512
#define BATCH   8
#define DDIM    4096
#define TOPK    512
#define KSTEPS  (N_EMB / 4)   // 128 WMMA k-steps

typedef __attribute__((ext_vector_type(2))) float v2f;
typedef __attribute__((ext_vector_type(8))) float v8f;

// ---------------------------------------------------------------------------
// K1: logits[b,d,e] = sum_h x[b,h,d] * W[e,h]  via V_WMMA_F32_16X16X4_F32
// One wave per 16(d) x 16(e-padded) tile, K looped over 512 in steps of 4.
// A (16x4 f32): lane M = lane&15, VGPR0/1 hold K = kb, kb+1, kb = 2*(lane>>4)
// B (4x16 f32): lane N = lane&15, VGPR0/1 hold K = kb, kb+1 (experts >=8 -> 0)
// C/D (16x16 f32, 8 VGPRs): lane N = lane&15, VGPR r -> M = r + 8*(lane>>4)
//
// B fragments for all 128 k-steps are staged once per block into LDS in the
// exact WMMA lane layout, so the hot loop is 2 global A-loads + 1 ds_load_b64
// + 1 v_wmma, with no exec-mask predication.
// ---------------------------------------------------------------------------
__global__ void logits_wmma_kernel(const float* __restrict__ x,
                                   const float* __restrict__ W,
                                   float* __restrict__ logits) {
    __shared__ v2f bfrag[KSTEPS * 32];   // 32 KB of 320 KB LDS

    const int tid  = threadIdx.x;
    const int lane = tid & 31;
    const int wave = tid >> 5;

    // Stage B fragments (branch-free: clamped load + cndmask select).
    for (int i = tid; i < KSTEPS * 32; i += 256) {
        const int s  = i >> 5;
        const int l  = i & 31;
        const int N  = l & 15;
        const int kb = (l >> 4) << 1;
        const bool ve = (N < NUM_EXP);
        const float* wrow = W + (ve ? N : 0) * N_EMB;   // always-valid address
        const float w0 = wrow[s * 4 + kb];
        const float w1 = wrow[s * 4 + kb + 1];
        v2f bm;
        bm.x = ve ? w0 : 0.0f;
        bm.y = ve ? w1 : 0.0f;
        bfrag[i] = bm;
    }
    __syncthreads();

    const int b  = blockIdx.y;
    const int d0 = (blockIdx.x * 8 + wave) * 16;
    const int M  = lane & 15;
    const int kb = (lane >> 4) << 1;     // 0 or 2
    const int hi = lane >> 4;

    // A pointer for this lane: x[b, kb, d0+M]; rows advance by DDIM floats.
    const float* xa = x + (size_t)b * N_EMB * DDIM + (size_t)kb * DDIM + d0 + M;
    const v2f* bp = bfrag + lane;

    v8f c = {};
    for (int s = 0; s < KSTEPS; ++s) {
        v2f a;
        a.x = xa[0];         // K = kb
        a.y = xa[DDIM];      // K = kb + 1
        const v2f bm = bp[s * 32];
        c = __builtin_amdgcn_wmma_f32_16x16x4_f32(
                /*neg_a=*/false, a, /*neg_b=*/false, bm,
                /*c_mod=*/(short)0, c, /*reuse_a=*/false, /*reuse_b=*/false);
        xa += 4 * DDIM;
    }

    if (M < NUM_EXP) {       // lane's N column is a real expert
#pragma unroll
        for (int r = 0; r < 8; ++r) {
            const int m = r + 8 * hi;
            logits[((size_t)b * DDIM + d0 + m) * NUM_EXP + M] = c[r];
        }
    }
}

// ---------------------------------------------------------------------------
// K2: probs_t[b][e][d] = softmax_e(logits[b][d][:]) + noise[b][d][e]
// ---------------------------------------------------------------------------
__global__ void softmax_noise_kernel(const float* __restrict__ logits,
                                     const float* __restrict__ noise,
                                     float* __restrict__ probs_t) {
    const int t = blockIdx.x * blockDim.x + threadIdx.x;
    if (t >= BATCH * DDIM) return;
    const int b = t >> 12;
    const int d = t & (DDIM - 1);

    float l[NUM_EXP];
    float mx = -3.402823466e38f;
#pragma unroll
    for (int e = 0; e < NUM_EXP; ++e) {
        l[e] = logits[(size_t)t * NUM_EXP + e];
        mx = fmaxf(mx, l[e]);
    }
    float s = 0.0f;
#pragma unroll
    for (int e = 0; e < NUM_EXP; ++e) {
        l[e] = expf(l[e] - mx);
        s += l[e];
    }
    const float inv = 1.0f / s;
#pragma unroll
    for (int e = 0; e < NUM_EXP; ++e) {
        probs_t[((size_t)(b * NUM_EXP + e)) * DDIM + d] =
            l[e] * inv + noise[(size_t)t * NUM_EXP + e];
    }
}

// ---------------------------------------------------------------------------
// K3: exact top-k per (b,e) row via rank counting (matches lax.top_k:
// descending values, ties -> lower index). One block (512 thr) per row.
// rank(d) = #{v_i > v_d} + #{v_i == v_d, i < d}; selected iff rank < TOPK,
// output slot = rank.
// ---------------------------------------------------------------------------
__global__ void topk_kernel(const float* __restrict__ probs_t,
                            float* __restrict__ w_out,
                            float* __restrict__ i_out,
                            int*   __restrict__ idx_ws,
                            float* __restrict__ counts) {
    __shared__ float row[DDIM];
    const int be = blockIdx.x;            // b*NUM_EXP + e
    const int b  = be / NUM_EXP;
    const int e  = be % NUM_EXP;
    const float* src = probs_t + (size_t)be * DDIM;

    for (int i = threadIdx.x; i < DDIM; i += blockDim.x) row[i] = src[i];
    __syncthreads();

    float vals[8];
    int   ds[8];
    int   rank[8];
#pragma unroll
    for (int j = 0; j < 8; ++j) {
        ds[j]   = threadIdx.x + j * 512;
        vals[j] = row[ds[j]];
        rank[j] = 0;
    }

    for (int i = 0; i < DDIM; ++i) {
        const float v = row[i];           // LDS broadcast (same addr all lanes)
#pragma unroll
        for (int j = 0; j < 8; ++j)
            rank[j] += (v > vals[j]) || ((v == vals[j]) && (i < ds[j]));
    }

#pragma unroll
    for (int j = 0; j < 8; ++j) {
        if (rank[j] < TOPK) {
            const size_t o = ((size_t)(e * BATCH + b)) * TOPK + rank[j];
            w_out[o]  = vals[j];
            i_out[o]  = (float)ds[j];     // tuple concatenation upcasts to f32
            idx_ws[o] = ds[j];
            atomicAdd(&counts[e * DDIM + ds[j]], 1.0f);
        }
    }
}

// ---------------------------------------------------------------------------
// K4: lb_loss = mean((counts/(B*TOPK+1e-9) - 1/E)^2) over E*D entries
// ---------------------------------------------------------------------------
__global__ void lbloss_kernel(const float* __restrict__ counts,
                              float* __restrict__ out) {
    __shared__ float red[256];
    const float invden = 1.0f / ((float)(BATCH * TOPK) + 1e-9f);
    float acc = 0.0f;
    for (int i = threadIdx.x; i < NUM_EXP * DDIM; i += 256) {
        const float u = counts[i] * invden - (1.0f / NUM_EXP);
        acc += u * u;
    }
    red[threadIdx.x] = acc;
    __syncthreads();
    for (int s = 128; s > 0; s >>= 1) {
        if (threadIdx.x < s) red[threadIdx.x] += red[threadIdx.x + s];
        __syncthreads();
    }
    if (threadIdx.x == 0) out[0] = red[0] / (float)(NUM_EXP * DDIM);
}

// ---------------------------------------------------------------------------
// K5: selected[e][b][h][k] = x[b][h][ idx[e][b][k] ]
// block = (e*B + b)*N_EMB + h, thread = k. Coalesced writes; gather reads
// hit L2 (x = 64 MB < 192 MB L2).
// ---------------------------------------------------------------------------
__global__ void gather_kernel(const float* __restrict__ x,
                              const int* __restrict__ idx_ws,
                              float* __restrict__ sel) {
    const int blk = blockIdx.x;
    const int h   = blk & (N_EMB - 1);
    const int eb  = blk >> 9;             // e*BATCH + b
    const int b   = eb & (BATCH - 1);
    const int k   = threadIdx.x;
    const int d   = idx_ws[(size_t)eb * TOPK + k];
    sel[(size_t)blk * TOPK + k] = x[((size_t)b * N_EMB + h) * DDIM + d];
}

// ---------------------------------------------------------------------------
extern "C" void kernel_launch(void* const* d_in, const int* in_sizes, int n_in,
                              void* d_out, int out_size, void* d_ws, size_t ws_size,
                              hipStream_t stream) {
    (void)in_sizes; (void)n_in; (void)out_size; (void)ws_size;

    const float* x     = (const float*)d_in[0];   // (8, 512, 4096)
    const float* W     = (const float*)d_in[1];   // (8, 512)
    const float* noise = (const float*)d_in[2];   // (8, 4096, 8)

    float* out   = (float*)d_out;
    float* sel   = out;                                   // 8*8*512*512
    float* w_out = sel + (size_t)NUM_EXP * BATCH * N_EMB * TOPK;
    float* i_out = w_out + (size_t)NUM_EXP * BATCH * TOPK;
    float* lb    = i_out + (size_t)NUM_EXP * BATCH * TOPK;

    float* logits  = (float*)d_ws;                        // 262144 f32
    float* probs_t = logits + (size_t)BATCH * DDIM * NUM_EXP;
    float* counts  = probs_t + (size_t)BATCH * NUM_EXP * DDIM; // 32768 f32
    int*   idx_ws  = (int*)(counts + (size_t)NUM_EXP * DDIM);  // 32768 i32

    hipMemsetAsync(counts, 0, (size_t)NUM_EXP * DDIM * sizeof(float), stream);

    // K1: 8 waves/block, 16 d-rows/wave -> 128 rows/block, 32 blocks per batch
    logits_wmma_kernel<<<dim3(DDIM / 128, BATCH), 256, 0, stream>>>(x, W, logits);

    softmax_noise_kernel<<<(BATCH * DDIM + 255) / 256, 256, 0, stream>>>(
        logits, noise, probs_t);

    topk_kernel<<<BATCH * NUM_EXP, 512, 0, stream>>>(
        probs_t, w_out, i_out, idx_ws, counts);

    lbloss_kernel<<<1, 256, 0, stream>>>(counts, lb);

    gather_kernel<<<NUM_EXP * BATCH * N_EMB, TOPK, 0, stream>>>(x, idx_ws, sel);
}